// VNStdFeature_2894807957693
// MI455X (gfx1250) — compile-verified
//
#include <hip/hip_runtime.h>
#include <math.h>

// ---------------------------------------------------------------------------
// VNStdFeature on MI455X (gfx1250, wave32).
//
// Roofline: ~4.5 GFLOP fp32 vs ~400 MB HBM traffic (3-pass fused) =>
// memory-bound (~17us @ 23.3 TB/s). Channel GEMMs (32x64, 16x32 over 393k
// columns) run on V_WMMA_F32_16X16X4_F32 so the VALU only does the
// norm/BN/Gram-Schmidt elementwise work. BN stats require global reductions
// => 3 compute passes + init:
//   k_init   : zero 96-float accumulator block in d_ws
//   k_stats1 : p1 = W1*x via WMMA, accumulate per-channel sum/sumsq of |p1|
//   k_stats2 : recompute z1 (block1), stage via LDS, p2 = W2*z1 via WMMA,
//              accumulate block-2 norm stats
//   k_final  : full chain, frame via Gram-Schmidt, x_std = x^T frame
// One wave per 16-point tile; WMMA C-layout keeps BN math lane-local.
// ---------------------------------------------------------------------------

typedef float v2f __attribute__((ext_vector_type(2)));
typedef float v8f __attribute__((ext_vector_type(8)));

#define EPSF    1e-6f
#define BN_EPSF 1e-5f

namespace {
constexpr int kB = 8, kC = 64, kC2 = 32, kC4 = 16, kN = 16384;
constexpr int kTiles = kN / 16;                 // 1024 tiles per batch
constexpr float kInvCnt = 1.0f / (float)(kB * kN);
}

__device__ inline v8f vzero8() {
  v8f r;
#pragma unroll
  for (int i = 0; i < 8; ++i) r[i] = 0.0f;
  return r;
}

// D = A(16x4,f32) * B(4x16,f32) + C  -- fp32 WMMA, K=4
__device__ inline v8f wmma_f32_k4(v2f a, v2f b, v8f c) {
  return __builtin_amdgcn_wmma_f32_16x16x4_f32(false, a, false, b, (short)0, c,
                                               false, false);
}

// ---- block-1 GEMMs: p = W1*x (and optionally d = D1*x), 16-point tile ------
// A layout (16x4 f32): lane<16 holds K={0,1}, lane>=16 holds K={2,3}, M=lane&15
// B layout (4x16 f32): same K split; N = lane&15
// C/D layout: lane<16 rows 0..7, lane>=16 rows 8..15, col = lane&15
template <bool WITH_D>
__device__ inline void gemm1(const float* __restrict__ xb,
                             const float* __restrict__ W1,
                             const float* __restrict__ D1,
                             int n0, int lane, v8f p[3][2], v8f d[3][2]) {
  const int lh = lane >> 4, ln = lane & 15, koff = lh * 2;
#pragma unroll
  for (int v = 0; v < 3; ++v)
#pragma unroll
    for (int ot = 0; ot < 2; ++ot) {
      p[v][ot] = vzero8();
      if (WITH_D) d[v][ot] = vzero8();
    }
  for (int v = 0; v < 3; ++v) {
    for (int ks = 0; ks < 16; ++ks) {
      const int kc0 = ks * 4 + koff;                     // channel row pair
      const float* xc = xb + ((size_t)kc0 * 3 + v) * kN + n0 + ln;
      v2f bf;
      bf.x = xc[0];
      bf.y = xc[3 * kN];
#pragma unroll
      for (int ot = 0; ot < 2; ++ot) {
        const int o = ot * 16 + ln;
        v2f aw;
        aw.x = W1[o * kC + kc0];
        aw.y = W1[o * kC + kc0 + 1];
        p[v][ot] = wmma_f32_k4(aw, bf, p[v][ot]);
        if (WITH_D) {
          v2f ad;
          ad.x = D1[o * kC + kc0];
          ad.y = D1[o * kC + kc0 + 1];
          d[v][ot] = wmma_f32_k4(ad, bf, d[v][ot]);
        }
      }
    }
  }
}

// ---- block-1 full (BN + VN-LeakyReLU) -> z1 staged into LDS [c2][v][16] ----
__device__ inline void vn_block1_to_lds(const float* __restrict__ xb,
                                        const float* __restrict__ W1,
                                        const float* __restrict__ D1,
                                        const float* __restrict__ g1,
                                        const float* __restrict__ b1,
                                        const float* __restrict__ acc,
                                        int n0, int lane, float* zsh) {
  const int lh = lane >> 4, ln = lane & 15;
  v8f p[3][2], d[3][2];
  gemm1<true>(xb, W1, D1, n0, lane, p, d);
#pragma unroll
  for (int ot = 0; ot < 2; ++ot) {
#pragma unroll
    for (int r = 0; r < 8; ++r) {
      const int o = ot * 16 + lh * 8 + r;
      const float p0 = p[0][ot][r], p1v = p[1][ot][r], p2v = p[2][ot][r];
      const float nrm = sqrtf(p0 * p0 + p1v * p1v + p2v * p2v) + EPSF;
      const float m   = acc[o] * kInvCnt;
      const float var = acc[32 + o] * kInvCnt - m * m;
      const float nbn = (nrm - m) * (1.0f / sqrtf(var + BN_EPSF)) * g1[o] + b1[o];
      const float f = nbn / nrm;
      const float pv[3] = {p0 * f, p1v * f, p2v * f};
      const float dv[3] = {d[0][ot][r], d[1][ot][r], d[2][ot][r]};
      const float dt  = pv[0] * dv[0] + pv[1] * dv[1] + pv[2] * dv[2];
      const float dsq = dv[0] * dv[0] + dv[1] * dv[1] + dv[2] * dv[2];
      // 0.2*p + 0.8*(p - (dot<0)*dot/(dsq+eps)*d)  ==  p - coef*d
      const float coef = (dt >= 0.0f) ? 0.0f : (0.8f * dt / (dsq + EPSF));
#pragma unroll
      for (int v = 0; v < 3; ++v)
        zsh[((o * 3 + v) << 4) + ln] = pv[v] - coef * dv[v];
    }
  }
  __syncthreads();
}

// ---- block-2 GEMMs from LDS-staged z1 --------------------------------------
template <bool WITH_D>
__device__ inline void gemm2(const float* __restrict__ zsh,
                             const float* __restrict__ W2,
                             const float* __restrict__ D2,
                             int lane, v8f p2[3], v8f d2[3]) {
  const int lh = lane >> 4, ln = lane & 15, koff = lh * 2;
#pragma unroll
  for (int v = 0; v < 3; ++v) {
    p2[v] = vzero8();
    if (WITH_D) d2[v] = vzero8();
  }
  for (int v = 0; v < 3; ++v) {
#pragma unroll
    for (int ks = 0; ks < 8; ++ks) {
      const int kc0 = ks * 4 + koff;
      v2f bf;
      bf.x = zsh[((kc0 * 3 + v) << 4) + ln];
      bf.y = zsh[(((kc0 + 1) * 3 + v) << 4) + ln];
      v2f aw;
      aw.x = W2[ln * kC2 + kc0];
      aw.y = W2[ln * kC2 + kc0 + 1];
      p2[v] = wmma_f32_k4(aw, bf, p2[v]);
      if (WITH_D) {
        v2f ad;
        ad.x = D2[ln * kC2 + kc0];
        ad.y = D2[ln * kC2 + kc0 + 1];
        d2[v] = wmma_f32_k4(ad, bf, d2[v]);
      }
    }
  }
}

// ============================ kernels =======================================

__global__ void k_init(float* __restrict__ acc) {
  const int i = threadIdx.x;
  if (i < 96) acc[i] = 0.0f;
}

__global__ __launch_bounds__(32) void k_stats1(const float* __restrict__ x,
                                               const float* __restrict__ W1,
                                               float* __restrict__ acc) {
  const int bb = blockIdx.x / kTiles;
  const int n0 = (blockIdx.x % kTiles) * 16;
  const int lane = threadIdx.x;
  const int lh = lane >> 4, ln = lane & 15;
  const float* xb = x + (size_t)bb * kC * 3 * kN;
  v8f p[3][2], dun[3][2];
  gemm1<false>(xb, W1, nullptr, n0, lane, p, dun);
#pragma unroll
  for (int ot = 0; ot < 2; ++ot) {
#pragma unroll
    for (int r = 0; r < 8; ++r) {
      const float p0 = p[0][ot][r], p1v = p[1][ot][r], p2v = p[2][ot][r];
      const float nrm = sqrtf(p0 * p0 + p1v * p1v + p2v * p2v) + EPSF;
      float s = nrm, sq = nrm * nrm;
#pragma unroll
      for (int m = 1; m < 16; m <<= 1) {  // reduce over 16 points (half-wave)
        s  += __shfl_xor(s, m, 32);
        sq += __shfl_xor(sq, m, 32);
      }
      if (ln == 0) {
        const int o = ot * 16 + lh * 8 + r;
        atomicAdd(&acc[o], s);
        atomicAdd(&acc[32 + o], sq);
      }
    }
  }
}

__global__ __launch_bounds__(32) void k_stats2(const float* __restrict__ x,
                                               const float* __restrict__ W1,
                                               const float* __restrict__ D1,
                                               const float* __restrict__ g1,
                                               const float* __restrict__ b1,
                                               const float* __restrict__ W2,
                                               float* __restrict__ acc) {
  __shared__ float zsh[kC2 * 3 * 16];
  const int bb = blockIdx.x / kTiles;
  const int n0 = (blockIdx.x % kTiles) * 16;
  const int lane = threadIdx.x;
  const int lh = lane >> 4, ln = lane & 15;
  const float* xb = x + (size_t)bb * kC * 3 * kN;
  vn_block1_to_lds(xb, W1, D1, g1, b1, acc, n0, lane, zsh);
  v8f p2[3], dun[3];
  gemm2<false>(zsh, W2, nullptr, lane, p2, dun);
#pragma unroll
  for (int r = 0; r < 8; ++r) {
    const float p0 = p2[0][r], p1v = p2[1][r], pv2 = p2[2][r];
    const float nrm = sqrtf(p0 * p0 + p1v * p1v + pv2 * pv2) + EPSF;
    float s = nrm, sq = nrm * nrm;
#pragma unroll
    for (int m = 1; m < 16; m <<= 1) {
      s  += __shfl_xor(s, m, 32);
      sq += __shfl_xor(sq, m, 32);
    }
    if (ln == 0) {
      const int c4 = lh * 8 + r;
      atomicAdd(&acc[64 + c4], s);
      atomicAdd(&acc[80 + c4], sq);
    }
  }
}

__global__ __launch_bounds__(32) void k_final(
    const float* __restrict__ x, const float* __restrict__ W1,
    const float* __restrict__ D1, const float* __restrict__ g1,
    const float* __restrict__ b1, const float* __restrict__ W2,
    const float* __restrict__ D2, const float* __restrict__ g2,
    const float* __restrict__ b2, const float* __restrict__ Wlin,
    const float* __restrict__ acc, float* __restrict__ out) {
  __shared__ float zsh[kC2 * 3 * 16];
  const int bb = blockIdx.x / kTiles;
  const int n0 = (blockIdx.x % kTiles) * 16;
  const int lane = threadIdx.x;
  const int lh = lane >> 4, ln = lane & 15;
  const float* xb = x + (size_t)bb * kC * 3 * kN;

  vn_block1_to_lds(xb, W1, D1, g1, b1, acc, n0, lane, zsh);
  v8f p2[3], d2[3];
  gemm2<true>(zsh, W2, D2, lane, p2, d2);

  // block-2 BN + VN-LeakyReLU (lane-local, 8 channels per lane)
  float z2v[3][8];
#pragma unroll
  for (int r = 0; r < 8; ++r) {
    const int c4 = lh * 8 + r;
    const float p0 = p2[0][r], p1v = p2[1][r], pv2 = p2[2][r];
    const float nrm = sqrtf(p0 * p0 + p1v * p1v + pv2 * pv2) + EPSF;
    const float m   = acc[64 + c4] * kInvCnt;
    const float var = acc[80 + c4] * kInvCnt - m * m;
    const float nbn = (nrm - m) * (1.0f / sqrtf(var + BN_EPSF)) * g2[c4] + b2[c4];
    const float f = nbn / nrm;
    const float pv[3] = {p0 * f, p1v * f, pv2 * f};
    const float dv[3] = {d2[0][r], d2[1][r], d2[2][r]};
    const float dt  = pv[0] * dv[0] + pv[1] * dv[1] + pv[2] * dv[2];
    const float dsq = dv[0] * dv[0] + dv[1] * dv[1] + dv[2] * dv[2];
    const float coef = (dt >= 0.0f) ? 0.0f : (0.8f * dt / (dsq + EPSF));
#pragma unroll
    for (int v = 0; v < 3; ++v) z2v[v][r] = pv[v] - coef * dv[v];
  }

  // z0 = Wlin * z2 : partial per half-wave, combine lane n with lane n+16
  float u[3][3];  // u[k][component]
  {
    float z0[2][3];
#pragma unroll
    for (int j = 0; j < 2; ++j)
#pragma unroll
      for (int v = 0; v < 3; ++v) {
        float t = 0.0f;
#pragma unroll
        for (int r = 0; r < 8; ++r) t += Wlin[j * kC4 + lh * 8 + r] * z2v[v][r];
        t += __shfl_xor(t, 16, 32);  // rows 0-7 + rows 8-15
        z0[j][v] = t;
      }
    // Gram-Schmidt frame
    const float l1 = sqrtf(z0[0][0] * z0[0][0] + z0[0][1] * z0[0][1] +
                           z0[0][2] * z0[0][2]) + EPSF;
#pragma unroll
    for (int v = 0; v < 3; ++v) u[0][v] = z0[0][v] / l1;
    const float dvu = z0[1][0] * u[0][0] + z0[1][1] * u[0][1] + z0[1][2] * u[0][2];
    float w2v[3];
#pragma unroll
    for (int v = 0; v < 3; ++v) w2v[v] = z0[1][v] - dvu * u[0][v];
    const float l2 = sqrtf(w2v[0] * w2v[0] + w2v[1] * w2v[1] + w2v[2] * w2v[2]) + EPSF;
#pragma unroll
    for (int v = 0; v < 3; ++v) u[1][v] = w2v[v] / l2;
    u[2][0] = u[0][1] * u[1][2] - u[0][2] * u[1][1];
    u[2][1] = u[0][2] * u[1][0] - u[0][0] * u[1][2];
    u[2][2] = u[0][0] * u[1][1] - u[0][1] * u[1][0];
  }

  const int n = n0 + ln;
  // frame output: frame[b, v, k, n] = u_k[v]; only lower half writes
  if (lh == 0) {
    float* fout = out + (size_t)kB * kC * 3 * kN + (size_t)bb * 9 * kN + n;
#pragma unroll
    for (int v = 0; v < 3; ++v)
#pragma unroll
      for (int k = 0; k < 3; ++k) fout[(v * 3 + k) * kN] = u[k][v];
  }
  // x_std[b,i,k,n] = sum_v x[b,i,v,n] * u_k[v]; halves split i-channels
  const int i0 = lh * 32;
  for (int i = i0; i < i0 + 32; ++i) {
    const float* xi = xb + (size_t)i * 3 * kN + n;
    const float x0 = xi[0], x1 = xi[kN], x2 = xi[2 * kN];
    float* ob = out + ((size_t)bb * kC + i) * 3 * kN + n;
#pragma unroll
    for (int k = 0; k < 3; ++k)
      ob[k * kN] = x0 * u[k][0] + x1 * u[k][1] + x2 * u[k][2];
  }
}

// ============================ launcher ======================================

extern "C" void kernel_launch(void* const* d_in, const int* in_sizes, int n_in,
                              void* d_out, int out_size, void* d_ws,
                              size_t ws_size, hipStream_t stream) {
  const float* x    = (const float*)d_in[0];
  const float* W1   = (const float*)d_in[1];
  const float* D1   = (const float*)d_in[2];
  const float* g1   = (const float*)d_in[3];
  const float* b1   = (const float*)d_in[4];
  const float* W2   = (const float*)d_in[5];
  const float* D2   = (const float*)d_in[6];
  const float* g2   = (const float*)d_in[7];
  const float* b2   = (const float*)d_in[8];
  const float* Wlin = (const float*)d_in[9];
  float* out = (float*)d_out;
  float* acc = (float*)d_ws;  // 96 floats: sum1[32] sumsq1[32] sum2[16] sumsq2[16]

  const dim3 grid(kB * kTiles);  // 8192 tiles, one wave32 each
  k_init<<<1, 128, 0, stream>>>(acc);
  k_stats1<<<grid, 32, 0, stream>>>(x, W1, acc);
  k_stats2<<<grid, 32, 0, stream>>>(x, W1, D1, g1, b1, W2, acc);
  k_final<<<grid, 32, 0, stream>>>(x, W1, D1, g1, b1, W2, D2, g2, b2, Wlin,
                                   acc, out);
}